// DiskHeaders_66537633349880
// MI455X (gfx1250) — compile-verified
//
#include <hip/hip_runtime.h>
#include <hip/hip_bf16.h>
#include <math.h>

// Problem constants (match reference setup_inputs)
#define BB     4
#define HH     512
#define WW     512
#define HWSZ   (HH * WW)
#define CTOT   129
#define NDESC  128
#define JJ     512          // junctions per batch
#define KK     1024         // top-k keypoints per batch
#define NKP    (JJ + KK)    // 1536 total keypoints per batch
#define WIN    7            // half of 15x15 window
#define THRESH 0.99f
#define CAND_MAX 8192
#define SORT_N   8192       // power of two >= CAND_MAX

typedef __attribute__((ext_vector_type(2))) float v2f;
typedef __attribute__((ext_vector_type(8))) float v8f;

// ---------------------------------------------------------------------------
// Kernel 1: separable NMS, horizontal pass. Also zeroes candidate counters.
// ---------------------------------------------------------------------------
__global__ void k_rowmax(const float* __restrict__ in,
                         float* __restrict__ rowmax,
                         int* __restrict__ counts) {
  int t = blockIdx.x * blockDim.x + threadIdx.x;
  if (t < BB) counts[t] = 0;          // reset counters (kernel order guarantees visibility)
  if (t >= BB * HWSZ) return;
  int b = t / HWSZ;
  int p = t - b * HWSZ;
  int y = p / WW;
  int x = p - y * WW;
  const float* hrow = in + (size_t)b * CTOT * HWSZ + (size_t)y * WW;
  // hint the next row into cache (speculative, dropped if it would fault)
  __builtin_prefetch(hrow + WW + x, 0, 1);
  int x0 = (x - WIN < 0) ? 0 : x - WIN;
  int x1 = (x + WIN > WW - 1) ? WW - 1 : x + WIN;
  float m = -__builtin_inff();
  for (int xx = x0; xx <= x1; ++xx) m = fmaxf(m, hrow[xx]);
  rowmax[t] = m;
}

// ---------------------------------------------------------------------------
// Kernel 2: vertical pass + threshold + sparse candidate compaction.
// ---------------------------------------------------------------------------
__global__ void k_colmax_cand(const float* __restrict__ in,
                              const float* __restrict__ rowmax,
                              float* __restrict__ cand_val,
                              int* __restrict__ cand_idx,
                              int* __restrict__ counts) {
  int t = blockIdx.x * blockDim.x + threadIdx.x;
  if (t >= BB * HWSZ) return;
  int b = t / HWSZ;
  int p = t - b * HWSZ;
  int y = p / WW;
  int x = p - y * WW;
  int y0 = (y - WIN < 0) ? 0 : y - WIN;
  int y1 = (y + WIN > HH - 1) ? HH - 1 : y + WIN;
  const float* rm = rowmax + (size_t)b * HWSZ;
  float m = -__builtin_inff();
  for (int yy = y0; yy <= y1; ++yy) m = fmaxf(m, rm[(size_t)yy * WW + x]);
  float v = in[(size_t)b * CTOT * HWSZ + p];  // channel 0 = heatmap
  if (v == m && v > THRESH) {
    int slot = atomicAdd(&counts[b], 1);
    if (slot < CAND_MAX) {
      cand_val[b * CAND_MAX + slot] = v;
      cand_idx[b * CAND_MAX + slot] = p;
    }
  }
}

// ---------------------------------------------------------------------------
// Kernel 3: per-batch exact top-K via bitonic sort in LDS (64 KB dynamic).
// Order: value descending, index ascending on ties (matches jax.lax.top_k).
// ---------------------------------------------------------------------------
__global__ void k_topk(const float* __restrict__ cand_val,
                       const int* __restrict__ cand_idx,
                       const int* __restrict__ counts,
                       float* __restrict__ kp_val,
                       int* __restrict__ kp_idx) {
  extern __shared__ char smem[];
  float* s_val = (float*)smem;
  int*   s_idx = (int*)(smem + SORT_N * sizeof(float));
  int b = blockIdx.x;
  int n = counts[b];
  if (n > CAND_MAX) n = CAND_MAX;
  for (int i = threadIdx.x; i < SORT_N; i += blockDim.x) {
    if (i < n) {
      s_val[i] = cand_val[b * CAND_MAX + i];
      s_idx[i] = cand_idx[b * CAND_MAX + i];
    } else {
      s_val[i] = -__builtin_inff();
      s_idx[i] = 0x7fffffff;
    }
  }
  __syncthreads();
  for (unsigned k = 2; k <= SORT_N; k <<= 1) {
    for (unsigned j = k >> 1; j > 0; j >>= 1) {
      for (unsigned t = threadIdx.x; t < SORT_N; t += blockDim.x) {
        unsigned q = t ^ j;
        if (q > t) {
          bool region_desc = ((t & k) == 0);
          float va = s_val[t], vb = s_val[q];
          int   ia = s_idx[t], ib = s_idx[q];
          // lt(a,b): a must strictly precede b in final (descending) order
          bool b_lt_a = (vb > va) || (vb == va && ib < ia);
          bool a_lt_b = (va > vb) || (va == vb && ia < ib);
          bool doswap = region_desc ? b_lt_a : a_lt_b;
          if (doswap) {
            s_val[t] = vb; s_val[q] = va;
            s_idx[t] = ib; s_idx[q] = ia;
          }
        }
      }
      __syncthreads();
    }
  }
  for (int i = threadIdx.x; i < KK; i += blockDim.x) {
    kp_val[b * KK + i] = s_val[i];
    kp_idx[b * KK + i] = s_idx[i];
  }
}

// ---------------------------------------------------------------------------
// Kernel 4: descriptor gather + L2 norm (sum-of-squares via V_WMMA_F32_16X16X4_F32)
// + normalized write-out + keypoints + scores.
// One wave handles 16 keypoint rows; 4 waves / block; grid = B*96/4 blocks.
// ---------------------------------------------------------------------------
__global__ void __launch_bounds__(128) k_desc(const float* __restrict__ in,
                                              const int* __restrict__ juncs,
                                              const float* __restrict__ kp_val,
                                              const int* __restrict__ kp_idx,
                                              float* __restrict__ out) {
  __shared__ float s_sum[4][16];
  __shared__ int   s_pix[4][16];

  int lane = threadIdx.x & 31;
  int wib  = threadIdx.x >> 5;            // wave in block
  int gw   = blockIdx.x * 4 + wib;        // global wave id
  int b    = gw / 96;
  int g    = gw - b * 96;                 // 16-row group within batch
  int m    = lane & 15;                   // row within group (A-matrix M)
  int h    = lane >> 4;                   // lane half (A-matrix K-pair select)
  int r    = g * 16 + m;                  // keypoint row 0..1535

  float maxv = kp_val[b * KK];            // sorted descending -> element 0 is the max
  int xk, yk;
  float score;
  if (r < JJ) {
    xk = juncs[((size_t)b * JJ + r) * 2 + 0];
    yk = juncs[((size_t)b * JJ + r) * 2 + 1];
    score = 1.0f;
  } else {
    int idx = kp_idx[b * KK + (r - JJ)];
    int ci = idx;
    if (ci < 0) ci = 0;
    if (ci > HWSZ - 1) ci = HWSZ - 1;     // clamp padded (-inf) slots for addressing
    xk = ci % WW;
    yk = ci / WW;
    score = kp_val[b * KK + (r - JJ)] / maxv;
  }
  int pix = yk * WW + xk;
  if (pix < 0) pix = 0;
  if (pix >= HWSZ) pix = HWSZ - 1;

  const float* fb = in + (size_t)b * CTOT * HWSZ + (size_t)HWSZ;  // fmap = channels 1..128

  // Pass 1: sum of squares for 16 rows via WMMA.
  // A[m][k] = x[m][k0+k]^2 (16x4 f32, ISA lane layout: h=0 -> K={0,1}, h=1 -> K={2,3}),
  // B = ones(4x16)  =>  D[m][n] = row sumsq, replicated over n.
  v8f acc = {};
  v2f bones;
  bones.x = 1.0f;
  bones.y = 1.0f;
  for (int k0 = 0; k0 < NDESC; k0 += 4) {
    int c = k0 + 2 * h;
    float x0 = fb[(size_t)c * HWSZ + pix];
    float x1 = fb[(size_t)(c + 1) * HWSZ + pix];
    v2f a;
    a.x = x0 * x0;
    a.y = x1 * x1;
    acc = __builtin_amdgcn_wmma_f32_16x16x4_f32(false, a, false, bones,
                                                (short)0, acc, false, false);
  }
  // C/D layout: lanes 0-15 hold M=0..7 in vgpr 0..7, lanes 16-31 hold M=8..15.
  if ((lane & 15) == 0) {
#pragma unroll
    for (int rr = 0; rr < 8; ++rr) s_sum[wib][h * 8 + rr] = acc[rr];
  }
  if (h == 0) s_pix[wib][m] = pix;
  __syncthreads();

  // Pass 2: normalize + coalesced write. Each lane owns 4 consecutive channels.
  const size_t desc_base = (size_t)BB * NKP * 2;
  int c0 = lane * 4;
  for (int mm = 0; mm < 16; ++mm) {
    int rr = g * 16 + mm;
    int pm = s_pix[wib][mm];
    float nrm = sqrtf(s_sum[wib][mm]);
    float inv = 1.0f / fmaxf(nrm, 1e-12f);
    float4 o;
    o.x = fb[(size_t)(c0 + 0) * HWSZ + pm] * inv;
    o.y = fb[(size_t)(c0 + 1) * HWSZ + pm] * inv;
    o.z = fb[(size_t)(c0 + 2) * HWSZ + pm] * inv;
    o.w = fb[(size_t)(c0 + 3) * HWSZ + pm] * inv;
    *(float4*)(out + desc_base + ((size_t)(b * NKP + rr)) * NDESC + c0) = o;
  }

  // keypoints_final + scores (one writer lane per row)
  if (h == 0) {
    out[((size_t)b * NKP + r) * 2 + 0] = (float)xk;
    out[((size_t)b * NKP + r) * 2 + 1] = (float)yk;
    out[(size_t)BB * NKP * 2 + (size_t)BB * NKP * NDESC + (size_t)b * NKP + r] = score;
  }
}

// ---------------------------------------------------------------------------
extern "C" void kernel_launch(void* const* d_in, const int* in_sizes, int n_in,
                              void* d_out, int out_size, void* d_ws, size_t ws_size,
                              hipStream_t stream) {
  const float* in    = (const float*)d_in[0];
  const int*   juncs = (const int*)d_in[1];
  float*       out   = (float*)d_out;

  // Workspace carve-up (~4.35 MB total)
  char* ws = (char*)d_ws;
  size_t off = 0;
  float* rowmax   = (float*)(ws + off); off += (size_t)BB * HWSZ * sizeof(float);
  float* cand_val = (float*)(ws + off); off += (size_t)BB * CAND_MAX * sizeof(float);
  int*   cand_idx = (int*)(ws + off);   off += (size_t)BB * CAND_MAX * sizeof(int);
  int*   counts   = (int*)(ws + off);   off += 256;
  float* kp_val   = (float*)(ws + off); off += (size_t)BB * KK * sizeof(float);
  int*   kp_idx   = (int*)(ws + off);   off += (size_t)BB * KK * sizeof(int);
  (void)ws_size; (void)in_sizes; (void)n_in; (void)out_size;

  int total = BB * HWSZ;
  k_rowmax<<<(total + 255) / 256, 256, 0, stream>>>(in, rowmax, counts);
  k_colmax_cand<<<(total + 255) / 256, 256, 0, stream>>>(in, rowmax, cand_val, cand_idx, counts);
  k_topk<<<BB, 1024, SORT_N * (sizeof(float) + sizeof(int)), stream>>>(cand_val, cand_idx, counts,
                                                                       kp_val, kp_idx);
  k_desc<<<(BB * 96) / 4, 128, 0, stream>>>(in, juncs, kp_val, kp_idx, out);
}